// FusionExtractor_28613072126538
// MI455X (gfx1250) — compile-verified
//
#include <hip/hip_runtime.h>
#include <cstdint>
#include <math.h>

namespace {
constexpr int GD     = 200;       // grid dim
constexpr int NSAMP  = 9;
constexpr int NFEAT  = 8;
constexpr int HWPix  = 512 * 512;
constexpr int BLK    = 256;

__device__ __forceinline__ const float* grid_row(const float* g, int x, int y, int z) {
    return g + ((((size_t)x * GD + y) * GD + z) << 3);   // 8 feats per cell, 32B aligned
}
}

__global__ __launch_bounds__(BLK)
void fusion_extract_kernel(const float* __restrict__ vpoints,
                           const float* __restrict__ veye,
                           const float* __restrict__ grid,
                           const float* __restrict__ empty_value,
                           float* __restrict__ out_values,
                           float* __restrict__ out_pn,
                           float* __restrict__ out_dirs)
{
    __shared__ __align__(16) float s_vp [BLK * 3];
    __shared__ __align__(16) float s_pn [BLK * NSAMP * 3];
    __shared__ __align__(16) float s_dir[BLK * 3];

    const int tid  = threadIdx.x;
    const int pix0 = blockIdx.x * BLK;
    const int pix  = pix0 + tid;

    // ---- CDNA5 async global->LDS stage of this block's vpoints (3072 B = 192 lanes x 16 B)
    if (tid < (BLK * 3) / 4) {
        uint32_t ldsOff = (uint32_t)(uintptr_t)(&s_vp[tid * 4]);
        const float* gsrc = vpoints + (size_t)pix0 * 3 + tid * 4;
        asm volatile("global_load_async_to_lds_b128 %0, %1, off"
                     :: "v"(ldsOff), "v"((uint64_t)(uintptr_t)gsrc)
                     : "memory");
    }
    asm volatile("s_wait_asynccnt 0x0" ::: "memory");
    __syncthreads();

    const float ex = veye[0], ey = veye[1], ez = veye[2];
    const float4 ev0 = ((const float4*)empty_value)[0];
    const float4 ev1 = ((const float4*)empty_value)[1];

    const float vx = s_vp[tid * 3 + 0];
    const float vy = s_vp[tid * 3 + 1];
    const float vz = s_vp[tid * 3 + 2];

    float dx = vx - ex, dy = vy - ey, dz = vz - ez;
    const float invLen = 1.0f / sqrtf(dx * dx + dy * dy + dz * dz);
    dx *= invLen; dy *= invLen; dz *= invLen;
    s_dir[tid * 3 + 0] = dx;
    s_dir[tid * 3 + 1] = dy;
    s_dir[tid * 3 + 2] = dz;

    // Speculative prefetch of the ray-span endpoints' grid lines (global_prefetch_b8)
    {
        const float ax = fmaf(-4.f, dx, vx), ay = fmaf(-4.f, dy, vy), az = fmaf(-4.f, dz, vz);
        const float bx = fmaf( 4.f, dx, vx), by = fmaf( 4.f, dy, vy), bz = fmaf( 4.f, dz, vz);
        const int iax = min(max((int)ax, 0), GD - 1);
        const int iay = min(max((int)ay, 0), GD - 1);
        const int iaz = min(max((int)az, 0), GD - 1);
        const int ibx = min(max((int)bx, 0), GD - 1);
        const int iby = min(max((int)by, 0), GD - 1);
        const int ibz = min(max((int)bz, 0), GD - 1);
        __builtin_prefetch(grid_row(grid, iax, iay, iaz), 0, 1);
        __builtin_prefetch(grid_row(grid, ibx, iby, ibz), 0, 1);
    }

    #pragma unroll
    for (int s = 0; s < NSAMP; ++s) {
        const float off = (float)(s - NSAMP / 2);
        const float cx = fmaf(off, dx, vx);
        const float cy = fmaf(off, dy, vy);
        const float cz = fmaf(off, dz, vz);

        // points_normalized = cv - (floor(cv)+0.5)
        s_pn[tid * 27 + s * 3 + 0] = cx - (floorf(cx) + 0.5f);
        s_pn[tid * 27 + s * 3 + 1] = cy - (floorf(cy) + 0.5f);
        s_pn[tid * 27 + s * 3 + 2] = cz - (floorf(cz) + 0.5f);

        const bool valid = (cx >= 0.f) && (cx < (float)GD) &&
                           (cy >= 0.f) && (cy < (float)GD) &&
                           (cz >= 0.f) && (cz < (float)GD);

        float r[NFEAT];
        if (valid) {
            // Replicate reference arithmetic exactly: p = cv + 1, base = floor(p)
            const float px = cx + 1.f, py = cy + 1.f, pz = cz + 1.f;
            const float fx = floorf(px), fy = floorf(py), fz = floorf(pz);
            const float tx = px - fx, ty = py - fy, tz = pz - fz;
            const int bxi = (int)fx, byi = (int)fy, bzi = (int)fz;
            // Edge-padded gp[base+shift] == grid[clamp(base-1+shift, 0, 199)]
            const int xi[2] = { min(max(bxi - 1, 0), GD - 1), min(bxi, GD - 1) };
            const int yi[2] = { min(max(byi - 1, 0), GD - 1), min(byi, GD - 1) };
            const int z0   =   min(max(bzi - 1, 0), GD - 1);
            const int z1   =   min(bzi, GD - 1);
            const float wx_[2] = { 1.f - tx, tx };
            const float wy_[2] = { 1.f - ty, ty };
            const float wz0 = 1.f - tz, wz1 = tz;

            float a[NFEAT] = {0.f, 0.f, 0.f, 0.f, 0.f, 0.f, 0.f, 0.f};
            #pragma unroll
            for (int i = 0; i < 2; ++i) {
                #pragma unroll
                for (int j = 0; j < 2; ++j) {
                    const float w  = wx_[i] * wy_[j];
                    const float w0 = w * wz0;
                    const float w1 = w * wz1;
                    const float* r0p = grid_row(grid, xi[i], yi[j], z0);
                    const float* r1p = grid_row(grid, xi[i], yi[j], z1);
                    const float4 c00 = ((const float4*)r0p)[0];
                    const float4 c01 = ((const float4*)r0p)[1];
                    const float4 c10 = ((const float4*)r1p)[0];
                    const float4 c11 = ((const float4*)r1p)[1];
                    a[0] = fmaf(w0, c00.x, fmaf(w1, c10.x, a[0]));
                    a[1] = fmaf(w0, c00.y, fmaf(w1, c10.y, a[1]));
                    a[2] = fmaf(w0, c00.z, fmaf(w1, c10.z, a[2]));
                    a[3] = fmaf(w0, c00.w, fmaf(w1, c10.w, a[3]));
                    a[4] = fmaf(w0, c01.x, fmaf(w1, c11.x, a[4]));
                    a[5] = fmaf(w0, c01.y, fmaf(w1, c11.y, a[5]));
                    a[6] = fmaf(w0, c01.z, fmaf(w1, c11.z, a[6]));
                    a[7] = fmaf(w0, c01.w, fmaf(w1, c11.w, a[7]));
                }
            }
            #pragma unroll
            for (int f = 0; f < NFEAT; ++f) r[f] = a[f];
        } else {
            r[0] = ev0.x; r[1] = ev0.y; r[2] = ev0.z; r[3] = ev0.w;
            r[4] = ev1.x; r[5] = ev1.y; r[6] = ev1.z; r[7] = ev1.w;
        }

        // values[(s*8+f)*HW + pix] : coalesced across lanes, NT to protect L2 for grid
        float* vb = out_values + (size_t)(s * NFEAT) * HWPix + pix;
        #pragma unroll
        for (int f = 0; f < NFEAT; ++f)
            __builtin_nontemporal_store(r[f], vb + (size_t)f * HWPix);
    }

    __syncthreads();

    // ---- CDNA5 async LDS->global streaming of points_normalized and dirs.
    //      No VGPR round-trip; NT temporal hint keeps these write-once outputs
    //      out of L2 so grid gathers keep the cache.
    {
        // s_pn: 256*27 floats = 1728 x 16B chunks = 6.75 rounds of 256 lanes
        const uint32_t ldsPn = (uint32_t)(uintptr_t)(&s_pn[0]);
        float* dst = out_pn + (size_t)pix0 * 27;
        #pragma unroll
        for (int k = 0; k < 7; ++k) {
            const int i = tid + k * BLK;                 // 16B-chunk index
            if (i < (BLK * 27) / 4) {
                const uint32_t lofs  = ldsPn + (uint32_t)i * 16u;
                const uint64_t gaddr = (uint64_t)(uintptr_t)(dst + (size_t)i * 4);
                asm volatile("global_store_async_from_lds_b128 %0, %1, off th:TH_STORE_NT"
                             :: "v"(gaddr), "v"(lofs) : "memory");
            }
        }
        // s_dir: 256*3 floats = 192 x 16B chunks
        const uint32_t ldsDir = (uint32_t)(uintptr_t)(&s_dir[0]);
        float* dd = out_dirs + (size_t)pix0 * 3;
        if (tid < (BLK * 3) / 4) {
            const uint32_t lofs  = ldsDir + (uint32_t)tid * 16u;
            const uint64_t gaddr = (uint64_t)(uintptr_t)(dd + (size_t)tid * 4);
            asm volatile("global_store_async_from_lds_b128 %0, %1, off th:TH_STORE_NT"
                         :: "v"(gaddr), "v"(lofs) : "memory");
        }
    }
    // S_ENDPGM implies wait-idle, but make the dependency explicit.
    asm volatile("s_wait_asynccnt 0x0" ::: "memory");
}

extern "C" void kernel_launch(void* const* d_in, const int* in_sizes, int n_in,
                              void* d_out, int out_size, void* d_ws, size_t ws_size,
                              hipStream_t stream) {
    const float* vpoints = (const float*)d_in[0];   // (1, HW, 3)
    const float* veye    = (const float*)d_in[1];   // (1, 3)
    // d_in[2] = depth: shape-only in the reference, unused
    const float* grid    = (const float*)d_in[3];   // (200,200,200,8)
    const float* empty   = (const float*)d_in[4];   // (8,)

    float* out        = (float*)d_out;
    float* out_values = out;                                        // 72*HW
    float* out_pn     = out + (size_t)NSAMP * NFEAT * HWPix;        // HW*9*3
    float* out_dirs   = out_pn + (size_t)HWPix * NSAMP * 3;         // HW*3

    hipLaunchKernelGGL(fusion_extract_kernel,
                       dim3(HWPix / BLK), dim3(BLK), 0, stream,
                       vpoints, veye, grid, empty, out_values, out_pn, out_dirs);
}